// Downsample_v2_6219112645099
// MI455X (gfx1250) — compile-verified
//
#include <hip/hip_runtime.h>
#include <cstdint>

// out[n, i, j] = 0.5 * in[n, 2i, 2j]   (Haar LL+LH+HL+HH)/4 collapses to this.
// in:  (16*64) images of 512x512 f32,  out: (16*64) images of 256x256 f32.
//
// Roofline: zero useful FLOPs after algebraic collapse; traffic = 512MB reads
// (even rows only) + 256MB writes = 768MB @ 23.3TB/s ~= 33us. Pure HBM stream.
// WMMA is intentionally absent: a matmul formulation would read 1GB and add
// matrix-core work without changing the memory bound.
//
// CDNA5-specific path used:
//  - GLOBAL_LOAD_ASYNC_TO_LDS_B128 (ASYNCcnt DMA, no VGPR round trip),
//    th:TH_LOAD_NT so the single-use 512MB input stream doesn't churn L2,
//  - s_wait_asynccnt 0 (same-wave producer/consumer, no s_barrier needed),
//  - stride-2 LDS reads: dword addr 2*lane + 64*m -> 32 unique even banks,
//    conflict-free; compiler fuses pairs into ds_load_2addr_stride64_b32,
//  - non-temporal b32 stores (write-once 256MB output; >> 192MB L2).

#define IH 512
#define IW 512
#define OW 256
#define ROWS_PER_BLOCK 8   // 8 waves of 32 per block

__global__ __launch_bounds__(256) void
haar_collapse_kernel(const float* __restrict__ in, float* __restrict__ out, int n_rows) {
    __shared__ float smem[ROWS_PER_BLOCK * IW];   // 16 KB per block

    const int lane = threadIdx.x & 31;
    const int wave = threadIdx.x >> 5;
    const int r    = blockIdx.x * ROWS_PER_BLOCK + wave;   // global output-row id
    if (r >= n_rows) return;

    const int img = r >> 8;        // 256 output rows per image
    const int i   = r & 255;       // output row within image

    const float* grow = in  + (size_t)img * (IH * IW) + (size_t)(2 * i) * IW;
    float*       orow = out + (size_t)img * (OW * OW) + (size_t)i * OW;

    // LDS byte offset of this wave's 512-float staging row.
    const uint32_t lds_base = (uint32_t)(uintptr_t)(&smem[wave * IW]);

    // Stage the even input row into LDS via CDNA5 async global->LDS DMA.
    // Each issue: 32 lanes x 16B = 512B; 4 issues cover the 2KB row.
    // Non-temporal: this line is read exactly once.
#pragma unroll
    for (int k = 0; k < 4; ++k) {
        const uint64_t gaddr = (uint64_t)(grow + k * 128 + lane * 4);
        const uint32_t loff  = lds_base + (uint32_t)((k * 128 + lane * 4) * sizeof(float));
        asm volatile("global_load_async_to_lds_b128 %0, %1, off th:TH_LOAD_NT"
                     :: "v"(loff), "v"(gaddr)
                     : "memory");
    }
    // Wait for our own async copies; no cross-wave sharing, so no s_barrier.
    asm volatile("s_wait_asynccnt 0x0" ::: "memory");

    const float* lrow = &smem[wave * IW];

    // Column de-interleave: lane produces j = lane + 32*m, m = 0..7.
    // LDS read addr = dword 2*j -> even banks only -> conflict-free across 32 lanes.
    // Stores: per m, wave writes 128B contiguous, non-temporal.
#pragma unroll
    for (int m = 0; m < 8; ++m) {
        const int j = lane + 32 * m;
        const float v = 0.5f * lrow[2 * j];
        __builtin_nontemporal_store(v, &orow[j]);
    }
}

extern "C" void kernel_launch(void* const* d_in, const int* in_sizes, int n_in,
                              void* d_out, int out_size, void* d_ws, size_t ws_size,
                              hipStream_t stream) {
    const float* x = (const float*)d_in[0];
    float* out     = (float*)d_out;

    const int images = in_sizes[0] / (IH * IW);      // 16*64 = 1024
    const int n_rows = images * (IH / 2);            // 262,144 output rows
    const int blocks = (n_rows + ROWS_PER_BLOCK - 1) / ROWS_PER_BLOCK;  // 32,768

    haar_collapse_kernel<<<blocks, 256, 0, stream>>>(x, out, n_rows);
}